// SplatGaussian2D_75522704933321
// MI455X (gfx1250) — compile-verified
//
#include <hip/hip_runtime.h>
#include <hip/hip_bf16.h>
#include <stdint.h>

// SplatGaussian2D forward: out[b,c] = sum_n wgt(b,n) * sigmoid(basis(b,n)·rgbsh[n,:,c])
// B=32768 rays, N=512 gaussians, 9 SH coeffs, 3 channels. fp32 throughout.

#define NRAYS     32768
#define NG_TOTAL  512
#define NG        256          // gaussians staged per LDS chunk
#define BLOCK     256
#define S_MINF    (1.0f/30.0f)
#define S_MAXF    (1.0f/0.75f)

__device__ __forceinline__ uint32_t lds_addr_of(const void* p) {
    // generic pointers to LDS carry the LDS byte offset in the low 32 bits
    return (uint32_t)(uintptr_t)p;
}

__device__ __forceinline__ float fast_rcp(float v) {
#if __has_builtin(__builtin_amdgcn_rcpf)
    return __builtin_amdgcn_rcpf(v);     // v_rcp_f32
#else
    return 1.0f / v;
#endif
}

__device__ __forceinline__ float fast_rsq(float v) {
#if __has_builtin(__builtin_amdgcn_rsqf)
    return __builtin_amdgcn_rsqf(v);     // v_rsq_f32
#else
    return rsqrtf(v);
#endif
}

__device__ __forceinline__ bool wave_any(bool p) {
#if __has_builtin(__builtin_amdgcn_ballot_w32)
    return __builtin_amdgcn_ballot_w32(p) != 0u;
#else
    return __any(p);
#endif
}

// Heavy path for one gaussian (runs only when some lane in the wave is inside 5 sigma).
__device__ __forceinline__ void splat_one(const float* __restrict__ s_par,
                                          const float* __restrict__ s_sh,
                                          int g, float vx, float vy,
                                          float d2, bool valid,
                                          float& acc0, float& acc1, float& acc2)
{
    if (!wave_any(valid)) return;
    const float4 p1 = *(const float4*)&s_par[g * 8 + 4];   // cosA, sinA, sigmoid(op)
    const float wgt = valid ? __expf(-d2) * p1.z : 0.0f;

    // sin/cos(theta) without atan2: theta = atan2(u, w), rotation preserves norms
    const float u  = p1.x * vx - p1.y * vy;
    const float wv = p1.y * vx + p1.x * vy;
    const float r2 = u * u + wv * wv;
    const float ri = (r2 > 0.0f) ? fast_rsq(r2) : 0.0f;
    const float s1 = u * ri;
    const float c1 = (r2 > 0.0f) ? wv * ri : 1.0f;
    // angle-addition recurrences for sin/cos(k*theta), k=2..4
    const float s2 = s1 * c1 + c1 * s1;
    const float c2 = c1 * c1 - s1 * s1;
    const float s3 = s2 * c1 + c2 * s1;
    const float c3 = c2 * c1 - s2 * s1;
    const float s4 = s3 * c1 + c3 * s1;
    const float c4 = c3 * c1 - s3 * s1;
    const float bas[9] = {1.f, s1, c1, s2, c2, s3, c3, s4, c4};

    const float* sh = &s_sh[g * 27];
    float z0 = 0.f, z1 = 0.f, z2 = 0.f;
#pragma unroll
    for (int m = 0; m < 9; ++m) {
        z0 = fmaf(bas[m], sh[m * 3 + 0], z0);
        z1 = fmaf(bas[m], sh[m * 3 + 1], z1);
        z2 = fmaf(bas[m], sh[m * 3 + 2], z2);
    }
    acc0 = fmaf(wgt, fast_rcp(1.f + __expf(-z0)), acc0);
    acc1 = fmaf(wgt, fast_rcp(1.f + __expf(-z1)), acc1);
    acc2 = fmaf(wgt, fast_rcp(1.f + __expf(-z2)), acc2);
}

__global__ __launch_bounds__(BLOCK)
void splat2d_fwd(const float* __restrict__ x,        // (B,2)
                 const float* __restrict__ rgbsh,    // (N,9,3)
                 const float* __restrict__ opacity,  // (N,)
                 const float* __restrict__ mu,       // (N,2)
                 const float* __restrict__ scale,    // (N,2)
                 const float* __restrict__ angle,    // (N,)
                 float* __restrict__ out)            // (B,3)
{
    __shared__ __align__(16) float s_sh[NG * 27];    // SH coeffs chunk (async-staged)
    __shared__ __align__(16) float s_par[NG * 8];    // cx,cy,Sx,Sy,cosA,sinA,sig(op),pad

    const int tid = threadIdx.x;
    const int ray = blockIdx.x * BLOCK + tid;
    const float px = x[2 * ray + 0];
    const float py = x[2 * ray + 1];

    float acc0 = 0.f, acc1 = 0.f, acc2 = 0.f;

    for (int chunk = 0; chunk < NG_TOTAL / NG; ++chunk) {
        __syncthreads();   // previous chunk fully consumed before restaging

        // ---- derived per-gaussian params (one gaussian per thread) ----
        {
            const int gg = chunk * NG + tid;
            const float2 muv = ((const float2*)mu)[gg];
            const float2 scv = ((const float2*)scale)[gg];
            const float gmx = tanhf(muv.x) * 1.05f;
            const float gmy = tanhf(muv.y) * 1.05f;
            const float Sx = fminf(fmaxf(scv.x, 0.f), 1.f) * (S_MAXF - S_MINF) + S_MINF;
            const float Sy = fminf(fmaxf(scv.y, 0.f), 1.f) * (S_MAXF - S_MINF) + S_MINF;
            const float al = tanhf(angle[gg]) * 3.1416f;
            float sa, ca;
            __sincosf(al, &sa, &ca);
            float* p = &s_par[tid * 8];
            p[0] = (gmx + 1.f) * 256.f;           // gaussian center in pixels (W/2=256)
            p[1] = (gmy + 1.f) * 256.f;
            p[2] = Sx;  p[3] = Sy;
            p[4] = ca;  p[5] = sa;
            p[6] = fast_rcp(1.f + __expf(-opacity[gg]));   // sigmoid(opacity)
            p[7] = 0.f;
        }

        // ---- CDNA5 async global->LDS copy of the SH chunk (27648 B) ----
        {
            const char*    src  = (const char*)(rgbsh + (size_t)chunk * NG * 27);
            const uint32_t base = lds_addr_of(s_sh);
            for (int i = tid; i < NG * 27 / 4; i += BLOCK) {      // 1728 b128 beats
                const uint32_t voff = (uint32_t)(i * 16);
                asm volatile("global_load_async_to_lds_b128 %0, %1, %2"
                             :: "v"(base + voff), "v"(voff), "s"(src)
                             : "memory");
            }
            asm volatile("s_wait_asynccnt 0" ::: "memory");
        }
        __syncthreads();

        // ---- accumulate over gaussians, 4 per trip to hide LDS latency ----
        for (int g = 0; g < NG; g += 4) {
            const float4 q0 = *(const float4*)&s_par[(g + 0) * 8];
            const float4 q1 = *(const float4*)&s_par[(g + 1) * 8];
            const float4 q2 = *(const float4*)&s_par[(g + 2) * 8];
            const float4 q3 = *(const float4*)&s_par[(g + 3) * 8];

            const float vx0 = px - q0.x, vy0 = py - q0.y;
            const float vx1 = px - q1.x, vy1 = py - q1.y;
            const float vx2 = px - q2.x, vy2 = py - q2.y;
            const float vx3 = px - q3.x, vy3 = py - q3.y;

            const float a0 = q0.z * vx0, b0 = q0.w * vy0;
            const float a1 = q1.z * vx1, b1 = q1.w * vy1;
            const float a2 = q2.z * vx2, b2 = q2.w * vy2;
            const float a3 = q3.z * vx3, b3 = q3.w * vy3;

            const float d0 = a0 * a0 + b0 * b0;
            const float d1 = a1 * a1 + b1 * b1;
            const float d2_ = a2 * a2 + b2 * b2;
            const float d3 = a3 * a3 + b3 * b3;

            const bool v0 = d0 < 25.0f;
            const bool v1 = d1 < 25.0f;
            const bool v2 = d2_ < 25.0f;
            const bool v3 = d3 < 25.0f;

            if (!wave_any(v0 | v1 | v2 | v3)) continue;   // whole group dead in wave

            splat_one(s_par, s_sh, g + 0, vx0, vy0, d0,  v0, acc0, acc1, acc2);
            splat_one(s_par, s_sh, g + 1, vx1, vy1, d1,  v1, acc0, acc1, acc2);
            splat_one(s_par, s_sh, g + 2, vx2, vy2, d2_, v2, acc0, acc1, acc2);
            splat_one(s_par, s_sh, g + 3, vx3, vy3, d3,  v3, acc0, acc1, acc2);
        }
    }

    out[3 * ray + 0] = acc0;
    out[3 * ray + 1] = acc1;
    out[3 * ray + 2] = acc2;
}

extern "C" void kernel_launch(void* const* d_in, const int* in_sizes, int n_in,
                              void* d_out, int out_size, void* d_ws, size_t ws_size,
                              hipStream_t stream) {
    (void)in_sizes; (void)n_in; (void)out_size; (void)d_ws; (void)ws_size;
    const float* x       = (const float*)d_in[0];
    const float* rgbsh   = (const float*)d_in[1];
    const float* opacity = (const float*)d_in[2];
    const float* mu      = (const float*)d_in[3];
    const float* scale   = (const float*)d_in[4];
    const float* angle   = (const float*)d_in[5];

    splat2d_fwd<<<NRAYS / BLOCK, BLOCK, 0, stream>>>(
        x, rgbsh, opacity, mu, scale, angle, (float*)d_out);
}